// JNet_25323127177528
// MI455X (gfx1250) — compile-verified
//
#include <hip/hip_runtime.h>
#include <cstdint>

// Problem constants (match reference)
#define B_    16
#define T_    400
#define J_    32
#define E_    256
#define TP1   401   // T+1
#define ANS_  2

typedef __attribute__((ext_vector_type(2))) float v2f;
typedef __attribute__((ext_vector_type(4))) float v4f;
typedef __attribute__((ext_vector_type(8))) float v8f;
typedef __attribute__((ext_vector_type(4))) unsigned int u32x4;
typedef __attribute__((ext_vector_type(4))) int i32x4;
typedef __attribute__((ext_vector_type(8))) int i32x8;

#define HSTR 260   // padded LDS row stride for 256-wide rows (64-bank conflict free)
#define GSTR 772   // padded LDS row stride for 768-wide rows

__device__ __forceinline__ float sigf(float x) { return 1.0f / (1.0f + expf(-x)); }

// ------------------------------------------------------------------
// Embedding gather (vectorized b128): EC = emb[context], EQ = emb[query]
// grid.x * 4 rows, 256 threads: 64 lanes x float4 per row
// ------------------------------------------------------------------
__global__ __launch_bounds__(256) void gather_embed_kernel(
    const int* __restrict__ ctx, const int* __restrict__ qry,
    const float* __restrict__ emb, float* __restrict__ EC,
    float* __restrict__ EQ) {
  int row = blockIdx.x * 4 + (threadIdx.x >> 6);
  int e4  = (threadIdx.x & 63) * 4;
  int tok;
  float* dst;
  if (row < B_ * T_) {
    tok = ctx[row];
    dst = EC + row * E_;
  } else {
    int r = row - B_ * T_;
    tok = qry[r];
    dst = EQ + r * E_;
  }
  *(v4f*)(dst + e4) = *(const v4f*)(emb + tok * E_ + e4);
}

// ------------------------------------------------------------------
// TDM: issue tensor_load_to_lds of a 16 x K fp32 tile (row stride lda)
// into LDS offset 0 with padded rows (stride HSTR dwords).
// D# encoding per CDNA5 ISA 8.3/8.4. Must be called by one wave only.
// (6-arg builtin form: g0, g1, g2, g3, g4(unused), cpol)
// ------------------------------------------------------------------
__device__ __forceinline__ void tdm_load_tile16(const float* gptr, int lda, int K) {
  uint64_t ga = (uint64_t)(uintptr_t)gptr;
  u32x4 g0;
  g0[0] = 1u;                                        // count=1, user mode
  g0[1] = 0u;                                        // lds_addr = 0
  g0[2] = (unsigned int)ga;                          // global_addr[31:0]
  g0[3] = (unsigned int)((ga >> 32) & 0x01FFFFFFu)   // global_addr[56:32]
          | (2u << 30);                              // type = 2 (image)
  i32x8 g1;
  g1[0] = (2 << 16)        // data_size = 4B
        | (1 << 20)        // pad_enable
        | (7 << 22)        // pad_interval: 256 dwords
        | (3 << 25);       // pad_amount: 4 dwords  -> LDS row stride 260
  g1[1] = (K & 0xFFFF) << 16;          // tensor_dim0[15:0]
  g1[2] = (16 << 16);                  // tensor_dim0[31:16]=0 | tensor_dim1=16
  g1[3] = (K & 0xFFFF) << 16;          // tensor_dim1 hi=0 | tile_dim0=K
  g1[4] = 16;                          // tile_dim1=16, tile_dim2=0
  g1[5] = lda;                         // tensor_dim0_stride[31:0]
  g1[6] = 0;
  g1[7] = 0;
  i32x4 g2 = {0, 0, 0, 0};
  i32x4 g3 = {0, 0, 0, 0};
  i32x8 g4 = {0, 0, 0, 0, 0, 0, 0, 0};
  __builtin_amdgcn_tensor_load_to_lds(g0, g1, g2, g3, g4, 0);
}

// ------------------------------------------------------------------
// Generic fp32 WMMA GEMM:  out[M,N] = A[M,K] * op(W) + bias + bias2
// One block (16 waves) per 16-row M tile: wave0 DMAs the A tile to LDS
// via TDM, each wave then computes N tiles tt = wave, wave+16, ...
// ------------------------------------------------------------------
template <bool TRANSB>
__global__ __launch_bounds__(512) void wmma_gemm_kernel(
    const float* __restrict__ A, int lda,
    const float* __restrict__ W, int ldw,
    const float* __restrict__ bias, const float* __restrict__ bias2,
    float* __restrict__ out, int ldo, int Ntiles, int K) {
  __shared__ float a_s[16 * HSTR];
  int tid = threadIdx.x;
  int lane = tid & 31, wave = tid >> 5;
  int m = lane & 15;       // A row / B col within tile
  int g = lane >> 4;       // lane-group (K half)
  int m0 = blockIdx.x * 16;

  if (tid < 32) {          // wave 0: async DMA of A tile, rows land at stride HSTR
    tdm_load_tile16(A + (size_t)m0 * lda, lda, K);
    __builtin_amdgcn_s_wait_tensorcnt(0);
  }
  __syncthreads();

  const float* hp = a_s + m * HSTR + g * 2;
  for (int tt = wave; tt < Ntiles; tt += 16) {
    int n0 = tt * 16;
    v8f acc = {};
    for (int k0 = 0; k0 < K; k0 += 16) {
      v2f a[4], b[4];
#pragma unroll
      for (int i = 0; i < 4; ++i) {
        int k = k0 + 4 * i;
        a[i].x = hp[k];  a[i].y = hp[k + 1];
        if (TRANSB) {
          const float* wp = W + (n0 + m) * ldw + k + g * 2;
          b[i].x = wp[0];  b[i].y = wp[1];
        } else {
          const float* wp = W + (k + g * 2) * ldw + (n0 + m);
          b[i].x = wp[0];  b[i].y = wp[ldw];
        }
      }
#pragma unroll
      for (int i = 0; i < 4; ++i)
        acc = __builtin_amdgcn_wmma_f32_16x16x4_f32(false, a[i], false, b[i],
                                                    (short)0, acc, false, false);
    }
    int n = n0 + m;
    float bb = 0.0f;
    if (bias)  bb += bias[n];
    if (bias2) bb += bias2[n];
#pragma unroll
    for (int v = 0; v < 8; ++v)
      out[(m0 + v + 8 * g) * ldo + n] = acc[v] + bb;
  }
}

// ------------------------------------------------------------------
// LSTM decoder scan: one persistent workgroup, 32 waves.
// gates(16x1024) = h(16x256) @ dec_Whh^T  (+ precomputed XG = x-gates+biases)
// XG layout: (B,T,1024)  row = b*T + t.  HR layout: (T+1,B,E) time-major.
// ------------------------------------------------------------------
__global__ __launch_bounds__(1024) void lstm_scan_kernel(
    const float* __restrict__ XG, const float* __restrict__ Whh,  // (1024,256)
    const float* __restrict__ h0, const float* __restrict__ c0,
    float* __restrict__ HR) {
  extern __shared__ float smem[];
  float* h_s = smem;               // 16 * HSTR
  float* g_s = smem + 16 * HSTR;   // 16 * 1024

  int tid  = threadIdx.x;
  int lane = tid & 31;
  int wave = tid >> 5;
  int m = lane & 15;
  int g = lane >> 4;

  for (int i = tid; i < B_ * E_; i += 1024) {
    int b = i >> 8, e = i & 255;
    h_s[b * HSTR + e] = h0[i];
    HR[i] = h0[i];                 // H_r[0] = h0
  }
  float c_reg[4];
#pragma unroll
  for (int j = 0; j < 4; ++j) c_reg[j] = c0[tid * 4 + j];
  __syncthreads();

  const float* hp = h_s + m * HSTR + g * 2;
  for (int t = 0; t < T_; ++t) {
    // gates GEMM: 64 column tiles over 32 waves, batched fragment loads
    for (int tt = wave; tt < 64; tt += 32) {
      int n0 = tt * 16;
      v8f acc = {};
      const float* wp = Whh + (n0 + m) * 256 + g * 2;
      for (int k0 = 0; k0 < 256; k0 += 16) {
        v2f a[4], b[4];
#pragma unroll
        for (int i = 0; i < 4; ++i) {
          int k = k0 + 4 * i;
          a[i].x = hp[k];  a[i].y = hp[k + 1];
          b[i].x = wp[k];  b[i].y = wp[k + 1];
        }
#pragma unroll
        for (int i = 0; i < 4; ++i)
          acc = __builtin_amdgcn_wmma_f32_16x16x4_f32(false, a[i], false, b[i],
                                                      (short)0, acc, false, false);
      }
      int n = n0 + m;
#pragma unroll
      for (int v = 0; v < 8; ++v) {
        int b = v + 8 * g;
        g_s[b * 1024 + n] = acc[v] + XG[(b * T_ + t) * 1024 + n];
      }
    }
    __syncthreads();

    // element-wise LSTM cell (gate order i,f,g,o); each thread owns 4 (b,e)
#pragma unroll
    for (int j = 0; j < 4; ++j) {
      int idx = tid * 4 + j;
      int b = idx >> 8, e = idx & 255;
      float gi = g_s[b * 1024 + e];
      float gf = g_s[b * 1024 + 256 + e];
      float gc = g_s[b * 1024 + 512 + e];
      float go = g_s[b * 1024 + 768 + e];
      float c2 = sigf(gf) * c_reg[j] + sigf(gi) * tanhf(gc);
      c_reg[j] = c2;
      float h2 = sigf(go) * tanhf(c2);
      h_s[b * HSTR + e] = h2;
      HR[(t + 1) * (B_ * E_) + idx] = h2;
    }
    __syncthreads();
  }
}

// ------------------------------------------------------------------
// GRU encoder scan (401 steps). GI = H_r @ enc_Wih^T + bih precomputed,
// layout (T+1, B, 768) row = t*16+b.  ENC layout (T+1,B,E).
// ------------------------------------------------------------------
__global__ __launch_bounds__(1024) void gru_scan_kernel(
    const float* __restrict__ GI, const float* __restrict__ Whh,  // (768,256)
    const float* __restrict__ bhh, float* __restrict__ ENC) {
  extern __shared__ float smem[];
  float* h_s = smem;               // 16 * HSTR
  float* g_s = smem + 16 * HSTR;   // 16 * GSTR

  int tid = threadIdx.x;
  int lane = tid & 31, wave = tid >> 5;
  int m = lane & 15, g = lane >> 4;

  for (int i = tid; i < B_ * E_; i += 1024) h_s[(i >> 8) * HSTR + (i & 255)] = 0.0f;
  __syncthreads();

  const float* hp = h_s + m * HSTR + g * 2;
  for (int t = 0; t < TP1; ++t) {
    for (int tt = wave; tt < 48; tt += 32) {
      int n0 = tt * 16;
      v8f acc = {};
      const float* wp = Whh + (n0 + m) * 256 + g * 2;
      for (int k0 = 0; k0 < 256; k0 += 16) {
        v2f a[4], b[4];
#pragma unroll
        for (int i = 0; i < 4; ++i) {
          int k = k0 + 4 * i;
          a[i].x = hp[k];  a[i].y = hp[k + 1];
          b[i].x = wp[k];  b[i].y = wp[k + 1];
        }
#pragma unroll
        for (int i = 0; i < 4; ++i)
          acc = __builtin_amdgcn_wmma_f32_16x16x4_f32(false, a[i], false, b[i],
                                                      (short)0, acc, false, false);
      }
      int n = n0 + m;
#pragma unroll
      for (int v = 0; v < 8; ++v)
        g_s[(v + 8 * g) * GSTR + n] = acc[v] + bhh[n];
    }
    __syncthreads();

#pragma unroll
    for (int j = 0; j < 4; ++j) {
      int idx = tid * 4 + j;
      int b = idx >> 8, e = idx & 255;
      const float* gi = GI + (t * B_ + b) * 768;
      float r  = sigf(gi[e] + g_s[b * GSTR + e]);
      float z  = sigf(gi[256 + e] + g_s[b * GSTR + 256 + e]);
      float nn = tanhf(gi[512 + e] + r * g_s[b * GSTR + 512 + e]);
      float hn = (1.0f - z) * nn + z * h_s[b * HSTR + e];
      h_s[b * HSTR + e] = hn;
      ENC[(t * B_ + b) * E_ + e] = hn;
    }
    __syncthreads();
  }
}

// ------------------------------------------------------------------
// G[b,t,j,e] = tanh(WHQ[b,j,e] + P[b,t,e] + R[t,b,e]) — b128 streaming
// ------------------------------------------------------------------
__global__ __launch_bounds__(256) void g_tanh_kernel(
    const float* __restrict__ WHQ, const float* __restrict__ P,
    const float* __restrict__ R, float* __restrict__ G) {
  int bt = blockIdx.x;            // b*T + t
  int b = bt / T_, t = bt % T_;
  int e4 = (threadIdx.x & 63) * 4;
  int j0 = threadIdx.x >> 6;      // 0..3
  v4f p = *(const v4f*)(P + bt * E_ + e4);
  v4f r = *(const v4f*)(R + (t * B_ + b) * E_ + e4);
  v4f s = p + r;
  const float* wq = WHQ + b * J_ * E_;
  float* gp = G + (size_t)bt * J_ * E_;
  for (int j = j0; j < J_; j += 4) {
    v4f w = *(const v4f*)(wq + j * E_ + e4);
    v4f o;
    o.x = tanhf(w.x + s.x);
    o.y = tanhf(w.y + s.y);
    o.z = tanhf(w.z + s.z);
    o.w = tanhf(w.w + s.w);
    __builtin_nontemporal_store(o, (v4f*)(gp + j * E_ + e4));
  }
}

// ------------------------------------------------------------------
// Pointer decoder: 2 GRU steps (dec_in == 0 -> gi = pdec_bih), additive
// attention scores over T+1, log_softmax -> probs (B,ANS,T+1).
// ------------------------------------------------------------------
__global__ __launch_bounds__(1024) void pointer_decoder_kernel(
    const float* __restrict__ BLEND,     // (T+1,B,E) row = t*16+b
    const float* __restrict__ Whh,       // pdec_Whh (768,256)
    const float* __restrict__ bih, const float* __restrict__ bhh,
    const float* __restrict__ W2,        // (256,256)
    const float* __restrict__ vt, float* __restrict__ probs) {
  extern __shared__ float smem[];
  float* h_s = smem;                     // 16*HSTR
  float* g_s = h_s + 16 * HSTR;          // 16*GSTR
  float* w_s = g_s + 16 * GSTR;          // 16*HSTR  (hidden @ W2^T)
  float* sc  = w_s + 16 * HSTR;          // 16*408   (scores)

  int tid = threadIdx.x;
  int lane = tid & 31, wave = tid >> 5;
  int m = lane & 15, g = lane >> 4;

  for (int i = tid; i < B_ * E_; i += 1024) h_s[(i >> 8) * HSTR + (i & 255)] = 0.0f;
  __syncthreads();

  const float* hp = h_s + m * HSTR + g * 2;
  for (int ans = 0; ans < ANS_; ++ans) {
    // gh = hidden @ pdec_Whh^T + bhh
    for (int tt = wave; tt < 48; tt += 32) {
      int n0 = tt * 16;
      v8f acc = {};
      const float* wp = Whh + (n0 + m) * 256 + g * 2;
      for (int k0 = 0; k0 < 256; k0 += 16) {
        v2f a[4], b[4];
#pragma unroll
        for (int i = 0; i < 4; ++i) {
          int k = k0 + 4 * i;
          a[i].x = hp[k];  a[i].y = hp[k + 1];
          b[i].x = wp[k];  b[i].y = wp[k + 1];
        }
#pragma unroll
        for (int i = 0; i < 4; ++i)
          acc = __builtin_amdgcn_wmma_f32_16x16x4_f32(false, a[i], false, b[i],
                                                      (short)0, acc, false, false);
      }
      int n = n0 + m;
#pragma unroll
      for (int v = 0; v < 8; ++v)
        g_s[(v + 8 * g) * GSTR + n] = acc[v] + bhh[n];
    }
    __syncthreads();

    // GRU cell with gi = bih (dec_in is all-zero in the reference)
#pragma unroll
    for (int j = 0; j < 4; ++j) {
      int idx = tid * 4 + j;
      int b = idx >> 8, e = idx & 255;
      float r  = sigf(bih[e] + g_s[b * GSTR + e]);
      float z  = sigf(bih[256 + e] + g_s[b * GSTR + 256 + e]);
      float nn = tanhf(bih[512 + e] + r * g_s[b * GSTR + 512 + e]);
      h_s[b * HSTR + e] = (1.0f - z) * nn + z * h_s[b * HSTR + e];
    }
    __syncthreads();

    // w2h = hidden @ W2^T  (16 tiles, waves 0..15)
    if (wave < 16) {
      int n0 = wave * 16;
      v8f acc = {};
      const float* wp = W2 + (n0 + m) * 256 + g * 2;
      for (int k0 = 0; k0 < 256; k0 += 16) {
        v2f a[4], b[4];
#pragma unroll
        for (int i = 0; i < 4; ++i) {
          int k = k0 + 4 * i;
          a[i].x = hp[k];  a[i].y = hp[k + 1];
          b[i].x = wp[k];  b[i].y = wp[k + 1];
        }
#pragma unroll
        for (int i = 0; i < 4; ++i)
          acc = __builtin_amdgcn_wmma_f32_16x16x4_f32(false, a[i], false, b[i],
                                                      (short)0, acc, false, false);
      }
      int n = n0 + m;
#pragma unroll
      for (int v = 0; v < 8; ++v) w_s[(v + 8 * g) * HSTR + n] = acc[v];
    }
    __syncthreads();

    // scores[b,t] = sum_e tanh(blend1[t,b,e] + w2h[b,e]) * vt[e]
    for (int p = tid; p < B_ * TP1; p += 1024) {
      int b = p / TP1, t = p % TP1;
      const float* bl = BLEND + (t * B_ + b) * E_;
      const float* wh = w_s + b * HSTR;
      float s = 0.0f;
      for (int e = 0; e < E_; ++e) s += tanhf(bl[e] + wh[e]) * vt[e];
      sc[b * 408 + t] = s;
    }
    __syncthreads();

    // log_softmax over t, one wave per batch row
    if (wave < 16) {
      int b = wave;
      float mx = -3.0e38f;
      for (int t = lane; t < TP1; t += 32) mx = fmaxf(mx, sc[b * 408 + t]);
      for (int o = 16; o > 0; o >>= 1) mx = fmaxf(mx, __shfl_xor(mx, o, 32));
      float sum = 0.0f;
      for (int t = lane; t < TP1; t += 32) sum += expf(sc[b * 408 + t] - mx);
      for (int o = 16; o > 0; o >>= 1) sum += __shfl_xor(sum, o, 32);
      float lse = mx + logf(sum);
      for (int t = lane; t < TP1; t += 32)
        probs[b * (ANS_ * TP1) + ans * TP1 + t] = sc[b * 408 + t] - lse;
    }
    __syncthreads();
  }
}

// ------------------------------------------------------------------
extern "C" void kernel_launch(void* const* d_in, const int* in_sizes, int n_in,
                              void* d_out, int out_size, void* d_ws, size_t ws_size,
                              hipStream_t stream) {
  (void)in_sizes; (void)n_in; (void)out_size; (void)ws_size;
  const int*   ctx      = (const int*)d_in[0];
  const int*   qry      = (const int*)d_in[1];
  const float* emb      = (const float*)d_in[2];
  const float* Wq       = (const float*)d_in[3];
  const float* Wp       = (const float*)d_in[4];
  const float* Wr       = (const float*)d_in[5];
  const float* dec_Wih  = (const float*)d_in[6];   // (1024,512)
  const float* dec_Whh  = (const float*)d_in[7];   // (1024,256)
  const float* dec_bih  = (const float*)d_in[8];
  const float* dec_bhh  = (const float*)d_in[9];
  const float* h0       = (const float*)d_in[10];
  const float* c0       = (const float*)d_in[11];
  const float* enc_Wih  = (const float*)d_in[12];  // (768,256)
  const float* enc_Whh  = (const float*)d_in[13];
  const float* enc_bih  = (const float*)d_in[14];
  const float* enc_bhh  = (const float*)d_in[15];
  const float* pdec_Whh = (const float*)d_in[17];
  const float* pdec_bih = (const float*)d_in[18];
  const float* pdec_bhh = (const float*)d_in[19];
  const float* W1       = (const float*)d_in[20];
  const float* W2       = (const float*)d_in[21];
  const float* vt       = (const float*)d_in[22];

  float* ws    = (float*)d_ws;
  float* EC    = ws;                        // 6400*256
  float* EQ    = EC + 6400 * 256;           // 512*256
  float* WHQ   = EQ + 512 * 256;            // 512*256
  float* P     = WHQ + 512 * 256;           // 6400*256
  float* XG    = P + 6400 * 256;            // 6400*1024
  float* HR    = XG + 6400 * 1024;          // 6416*256
  float* R     = HR + 6416 * 256;           // 6400*256
  float* GI    = R + 6400 * 256;            // 6416*768
  float* ENC   = GI + 6416 * 768;           // 6416*256
  float* BLEND = ENC + 6416 * 256;          // 6416*256

  float* probs = (float*)d_out;             // 16*2*401
  float* G     = probs + B_ * ANS_ * TP1;   // 16*400*32*256

  const size_t LSTM_SMEM = (16 * HSTR + 16 * 1024) * sizeof(float);
  const size_t GRU_SMEM  = (16 * HSTR + 16 * GSTR) * sizeof(float);
  const size_t DEC_SMEM  = (16 * HSTR + 16 * GSTR + 16 * HSTR + 16 * 408) * sizeof(float);
  (void)hipFuncSetAttribute((const void*)lstm_scan_kernel,
                            hipFuncAttributeMaxDynamicSharedMemorySize, (int)LSTM_SMEM);
  (void)hipFuncSetAttribute((const void*)gru_scan_kernel,
                            hipFuncAttributeMaxDynamicSharedMemorySize, (int)GRU_SMEM);
  (void)hipFuncSetAttribute((const void*)pointer_decoder_kernel,
                            hipFuncAttributeMaxDynamicSharedMemorySize, (int)DEC_SMEM);

  // 1) embeddings (6912 rows / 4 per block)
  gather_embed_kernel<<<(B_ * T_ + B_ * J_) / 4, 256, 0, stream>>>(ctx, qry, emb, EC, EQ);
  // 2) WHQ = EQ @ Wq            (M=512)
  wmma_gemm_kernel<false><<<32, 512, 0, stream>>>(EQ, 256, Wq, 256, nullptr, nullptr, WHQ, 256, 16, 256);
  // 3) P = EC @ Wp              (M=6400)
  wmma_gemm_kernel<false><<<400, 512, 0, stream>>>(EC, 256, Wp, 256, nullptr, nullptr, P, 256, 16, 256);
  // 4) XG = EC @ dec_Wih[:, :E]^T + bih + bhh
  wmma_gemm_kernel<true><<<400, 512, 0, stream>>>(EC, 256, dec_Wih, 512, dec_bih, dec_bhh, XG, 1024, 64, 256);
  // 5) LSTM scan -> HR (T+1,B,E)
  lstm_scan_kernel<<<1, 1024, LSTM_SMEM, stream>>>(XG, dec_Whh, h0, c0, HR);
  // 6) R = HR[0:T] @ Wr
  wmma_gemm_kernel<false><<<400, 512, 0, stream>>>(HR, 256, Wr, 256, nullptr, nullptr, R, 256, 16, 256);
  // 7) GI = HR @ enc_Wih^T + enc_bih   (M=6416)
  wmma_gemm_kernel<true><<<401, 512, 0, stream>>>(HR, 256, enc_Wih, 256, enc_bih, nullptr, GI, 768, 48, 256);
  // 8) GRU encoder scan -> ENC
  gru_scan_kernel<<<1, 1024, GRU_SMEM, stream>>>(GI, enc_Whh, enc_bhh, ENC);
  // 9) blend1 = ENC @ W1^T
  wmma_gemm_kernel<true><<<401, 512, 0, stream>>>(ENC, 256, W1, 256, nullptr, nullptr, BLEND, 256, 16, 256);
  // 10) G tensor (200 MB streaming write)
  g_tanh_kernel<<<B_ * T_, 256, 0, stream>>>(WHQ, P, R, G);
  // 11) pointer decoder -> probs
  pointer_decoder_kernel<<<1, 1024, DEC_SMEM, stream>>>(BLEND, pdec_Whh, pdec_bih, pdec_bhh, W2, vt, probs);
}